// DetectionNet_34110630265392
// MI455X (gfx1250) — compile-verified
//
#include <hip/hip_runtime.h>
#include <hip/hip_bf16.h>
#include <stdint.h>

// ---------------------------------------------------------------------------
// SSD detection post-processing for MI455X (gfx1250, wave32).
//
// Pipeline:
//   K1 decode_softmax : decode boxes (xyxy) + softmax probs, class-major store
//   K2 nms_per_class  : 80 blocks = (image,class); boxes staged to LDS via the
//                       Tensor Data Mover (tensor_load_to_lds + s_wait_tensorcnt),
//                       greedy NMS = repeated block-argmax + IoU suppression
//   K3 topk_write     : per-image top-200 over <=4000 LDS candidates + outputs
// ---------------------------------------------------------------------------

constexpr int NB   = 4;     // batch
constexpr int NP   = 3106;  // priors
constexpr int NCC  = 21;    // classes incl. background
constexpr int NCLS = 20;    // foreground classes
constexpr int TOPK = 200;
#define MIN_SCORE_F   0.05f
#define MAX_OVERLAP_F 0.45f
#define NEG_INF (-__builtin_huge_valf())

// workspace layout (float units)
constexpr size_t WS_BOXES  = 0;                                   // NB*NP*4 f32 (xyxy)
constexpr size_t WS_CLS    = WS_BOXES + (size_t)NB * NP * 4;      // NB*NCLS*NP f32 (class-major)
constexpr size_t WS_KSCORE = WS_CLS   + (size_t)NB * NCLS * NP;   // NB*NCLS*TOPK f32
constexpr size_t WS_KIDX   = WS_KSCORE + (size_t)NB * NCLS * TOPK;// NB*NCLS*TOPK i32
constexpr size_t WS_KCNT   = WS_KIDX   + (size_t)NB * NCLS * TOPK;// NB*NCLS i32

typedef unsigned int u32x4 __attribute__((ext_vector_type(4)));
typedef int          i32x8 __attribute__((ext_vector_type(8)));
typedef int          i32x4 __attribute__((ext_vector_type(4)));

// ---------------------------------------------------------------------------
// K1: decode + softmax.  One thread per (b,p).  ~12.4K threads, fully
// bandwidth-trivial (1.3 MB total), so simplicity wins.
// ---------------------------------------------------------------------------
__global__ __launch_bounds__(256) void decode_softmax(
    const float* __restrict__ locs, const float* __restrict__ scores,
    const float* __restrict__ priors, float* __restrict__ ws) {
  int t = blockIdx.x * blockDim.x + threadIdx.x;
  if (t >= NB * NP) return;
  int b = t / NP, p = t % NP;

  float4 l  = ((const float4*)locs)[t];
  float4 pr = ((const float4*)priors)[p];
  float cx = l.x * pr.z * 0.1f + pr.x;
  float cy = l.y * pr.w * 0.1f + pr.y;
  float w  = expf(l.z * 0.2f) * pr.z;
  float h  = expf(l.w * 0.2f) * pr.w;
  ((float4*)(ws + WS_BOXES))[t] =
      make_float4(cx - 0.5f * w, cy - 0.5f * h, cx + 0.5f * w, cy + 0.5f * h);

  const float* s = scores + (size_t)t * NCC;
  float m = s[0];
#pragma unroll
  for (int c = 1; c < NCC; ++c) m = fmaxf(m, s[c]);
  float e[NCC], sum = 0.f;
#pragma unroll
  for (int c = 0; c < NCC; ++c) { e[c] = expf(s[c] - m); sum += e[c]; }
  float inv = 1.0f / sum;
  float* cls = ws + WS_CLS + (size_t)b * NCLS * NP;   // class-major for K2 coalescing
#pragma unroll
  for (int c = 0; c < NCLS; ++c) cls[(size_t)c * NP + p] = e[c + 1] * inv;
}

// ---------------------------------------------------------------------------
// K2: per-(image,class) greedy NMS.  Boxes live in LDS (49.7 KB tile, DMA'd
// by the TDM).  Greedy selection == sorted-order NMS: each iteration keeps
// the argmax of remaining scores, then suppresses IoU > 0.45 in parallel.
// ---------------------------------------------------------------------------
__global__ __launch_bounds__(256) void nms_per_class(float* __restrict__ ws) {
  const int b = blockIdx.x / NCLS;
  const int c = blockIdx.x % NCLS;
  const int tid = threadIdx.x;

  __shared__ float4 sBox[NP];      // 49,696 B
  __shared__ float  sScore[NP];    // 12,424 B
  __shared__ float  redV[256];
  __shared__ int    redI[256];

  const float4* gBoxes = (const float4*)(ws + WS_BOXES) + (size_t)b * NP;

#if defined(__gfx1250__) && __has_builtin(__builtin_amdgcn_tensor_load_to_lds) && \
    __has_builtin(__builtin_amdgcn_s_wait_tensorcnt)
  // --- Tensor Data Mover: 1-D tile, 12424 x 4B elements, global -> LDS ---
  if (tid < 32) {                              // one wave issues the DMA
    unsigned long long ga = (unsigned long long)(size_t)gBoxes;
    // flat shared address truncated to 32b == LDS byte offset (ISA 10.2)
    unsigned int la = (unsigned int)(size_t)(void*)&sBox[0];
    u32x4 g0 = { 1u,                            // count=1 valid descriptor
                 la,                            // lds_addr
                 (unsigned)(ga & 0xFFFFFFFFu),  // global_addr[31:0]
                 (unsigned)((ga >> 32) & 0x01FFFFFFu) | (2u << 30) }; // [56:32] | type=2
    i32x8 g1 = { (int)(2u << 16),                       // data_size=4B
                 (int)((unsigned)(NP * 4) << 16),       // tensor_dim0 lo16 @ [31:16]
                 (int)(1 << 16),                        // dim0 hi16=0 | tensor_dim1=1
                 (int)((unsigned)(NP * 4) << 16),       // dim1 hi16=0 | tile_dim0 @ [31:16]
                 0,                                     // tile_dim1=0, tile_dim2=0
                 (int)(NP * 4),                         // tensor_dim0_stride lo32
                 0, 0 };
    i32x4 gz4 = { 0, 0, 0, 0 };
    i32x8 gz8 = { 0, 0, 0, 0, 0, 0, 0, 0 };
    // 6-arg form (clang-23 / therock-10.0): (g0, g1, g2, g3, g4, cpol)
    __builtin_amdgcn_tensor_load_to_lds(g0, g1, gz4, gz4, gz8, 0);
    __builtin_amdgcn_s_wait_tensorcnt((short)0);        // s_wait_tensorcnt 0
  }
#else
  for (int p = tid; p < NP; p += 256) sBox[p] = gBoxes[p];
#endif

  const float* clsRow = ws + WS_CLS + (size_t)(b * NCLS + c) * NP;
  __builtin_prefetch(clsRow + tid, 0, 0);               // global_prefetch_b8
  for (int p = tid; p < NP; p += 256) {
    float v = clsRow[p];
    sScore[p] = (v > MIN_SCORE_F) ? v : NEG_INF;
  }
  __syncthreads();

  float* kScore = ws + WS_KSCORE + (size_t)(b * NCLS + c) * TOPK;
  int*   kIdx   = (int*)(ws + WS_KIDX) + (size_t)(b * NCLS + c) * TOPK;
  int*   kCnt   = (int*)(ws + WS_KCNT);

  int kept = 0;
  while (kept < TOPK) {
    // block argmax, tie-break lowest index (matches stable argsort ordering)
    float bv = NEG_INF; int bi = -1;
    for (int p = tid; p < NP; p += 256) {
      float v = sScore[p];
      if (v > bv || (v == bv && (unsigned)p < (unsigned)bi)) { bv = v; bi = p; }
    }
    redV[tid] = bv; redI[tid] = bi;
    __syncthreads();
    for (int s = 128; s > 0; s >>= 1) {
      if (tid < s) {
        float ov = redV[tid + s]; int oi = redI[tid + s];
        if (ov > redV[tid] || (ov == redV[tid] && (unsigned)oi < (unsigned)redI[tid])) {
          redV[tid] = ov; redI[tid] = oi;
        }
      }
      __syncthreads();
    }
    float mv = redV[0]; int mi = redI[0];
    __syncthreads();
    if (!(mv > NEG_INF)) break;                         // uniform exit

    if (tid == 0) { kScore[kept] = mv; kIdx[kept] = mi; }
    float4 kb = sBox[mi];
    float ka = (kb.z - kb.x) * (kb.w - kb.y);
    for (int p = tid; p < NP; p += 256) {
      if (sScore[p] == NEG_INF) continue;
      float4 bb = sBox[p];
      float ix = fminf(kb.z, bb.z) - fmaxf(kb.x, bb.x);
      float iy = fminf(kb.w, bb.w) - fmaxf(kb.y, bb.y);
      float inter = fmaxf(ix, 0.f) * fmaxf(iy, 0.f);
      float ba = (bb.z - bb.x) * (bb.w - bb.y);
      float iou = inter / (ka + ba - inter);
      if (p == mi || iou > MAX_OVERLAP_F) sScore[p] = NEG_INF;
    }
    kept++;
    __syncthreads();
  }
  if (tid == 0) kCnt[b * NCLS + c] = kept;
}

// ---------------------------------------------------------------------------
// K3: per-image top-200 over class-capped candidates (<=4000), then write
// boxes / labels / scores / counts in the reference's concatenated layout.
// ---------------------------------------------------------------------------
__global__ __launch_bounds__(256) void topk_write(float* __restrict__ ws,
                                                  float* __restrict__ outF,
                                                  int* __restrict__ outI) {
  const int b = blockIdx.x, tid = threadIdx.x;
  constexpr int NC = NCLS * TOPK;                        // 4000 candidates
  __shared__ float sSc[NC];
  __shared__ int   sPid[NC];
  __shared__ float redV[256];
  __shared__ int   redI[256];
  __shared__ int   sTot;

  const float* kScore = ws + WS_KSCORE + (size_t)b * NCLS * TOPK;
  const int*   kIdx   = (const int*)(ws + WS_KIDX) + (size_t)b * NCLS * TOPK;
  const int*   kCnt   = (const int*)(ws + WS_KCNT) + b * NCLS;

  int localTot = 0;
  for (int t = tid; t < NC; t += 256) {
    int c = t / TOPK, j = t - c * TOPK;
    if (j < kCnt[c]) { sSc[t] = kScore[t]; sPid[t] = kIdx[t]; localTot++; }
    else             { sSc[t] = NEG_INF; }
  }
  redI[tid] = localTot;
  __syncthreads();
  for (int s = 128; s > 0; s >>= 1) {
    if (tid < s) redI[tid] += redI[tid + s];
    __syncthreads();
  }
  if (tid == 0) sTot = redI[0];
  __syncthreads();
  const int total = sTot;
  __syncthreads();

  const float4* gBoxes = (const float4*)(ws + WS_BOXES) + (size_t)b * NP;
  float* oBox = outF + (size_t)b * TOPK * 4;                         // [0, 3200)
  int*   oLab = outI + NB * TOPK * 4 + b * TOPK;                     // [3200, 4000)
  float* oSc  = outF + NB * TOPK * 4 + NB * TOPK + (size_t)b * TOPK; // [4000, 4800)
  int*   oCnt = outI + NB * TOPK * 4 + NB * TOPK * 2 + b;            // [4800, 4804)

  for (int i = 0; i < TOPK; ++i) {
    float bv = NEG_INF; int bi = -1;
    for (int t = tid; t < NC; t += 256) {
      float v = sSc[t];
      if (v > bv || (v == bv && (unsigned)t < (unsigned)bi)) { bv = v; bi = t; }
    }
    redV[tid] = bv; redI[tid] = bi;
    __syncthreads();
    for (int s = 128; s > 0; s >>= 1) {
      if (tid < s) {
        float ov = redV[tid + s]; int oi = redI[tid + s];
        if (ov > redV[tid] || (ov == redV[tid] && (unsigned)oi < (unsigned)redI[tid])) {
          redV[tid] = ov; redI[tid] = oi;
        }
      }
      __syncthreads();
    }
    float mv = redV[0]; int mi = redI[0];
    __syncthreads();
    if (tid == 0) {
      if (mv > NEG_INF) {
        float4 bx = gBoxes[sPid[mi]];
        oBox[i * 4 + 0] = bx.x; oBox[i * 4 + 1] = bx.y;
        oBox[i * 4 + 2] = bx.z; oBox[i * 4 + 3] = bx.w;
        oLab[i] = mi / TOPK + 1;
        oSc[i]  = mv;
        sSc[mi] = NEG_INF;
      } else {
        oBox[i * 4 + 0] = 0.f; oBox[i * 4 + 1] = 0.f;
        oBox[i * 4 + 2] = 0.f; oBox[i * 4 + 3] = 0.f;
        oLab[i] = 0;
        oSc[i]  = 0.f;
      }
    }
    __syncthreads();
  }
  if (tid == 0) {
    int cnt = total < TOPK ? total : TOPK;
    if (total == 0) { oBox[0] = 0.f; oBox[1] = 0.f; oBox[2] = 1.f; oBox[3] = 1.f; cnt = 1; }
    *oCnt = cnt;
  }
}

// ---------------------------------------------------------------------------
extern "C" void kernel_launch(void* const* d_in, const int* in_sizes, int n_in,
                              void* d_out, int out_size, void* d_ws, size_t ws_size,
                              hipStream_t stream) {
  const float* locs   = (const float*)d_in[0];   // (4, 3106, 4)
  const float* scores = (const float*)d_in[1];   // (4, 3106, 21)
  const float* priors = (const float*)d_in[2];   // (3106, 4)
  float* ws   = (float*)d_ws;
  float* outF = (float*)d_out;
  int*   outI = (int*)d_out;

  decode_softmax<<<(NB * NP + 255) / 256, 256, 0, stream>>>(locs, scores, priors, ws);
  nms_per_class<<<NB * NCLS, 256, 0, stream>>>(ws);
  topk_write<<<NB, 256, 0, stream>>>(ws, outF, outI);
}